// Mosaic_69492570849438
// MI455X (gfx1250) — compile-verified
//
#include <hip/hip_runtime.h>
#include <cstdint>

// Mosaic GEMM for MI455X (gfx1250, wave32):
//   y[M=8192, N=4096] = x[M, K=4096] @ W[K, N] + bias
//   W[kb*128 + r, nb*128 + c] = tiles[parts[kb, nb]][r][c]
// fp32 WMMA (V_WMMA_F32_16X16X4_F32) for reference-precision accumulation.
// Double-buffered GLOBAL_LOAD_ASYNC_TO_LDS_B128 staging (ASYNCcnt-tracked);
// all hot-loop LDS/global accesses use immediate offsets off hoisted bases.

typedef __attribute__((ext_vector_type(2))) float v2f;
typedef __attribute__((ext_vector_type(8))) float v8f;

#define BM 128
#define BN 128
#define BK 128
#define XSTRIDE (BK + 4)   // 132 dwords: lane-strided A reads hit distinct banks
#define K_DIM 4096
#define N_DIM 4096
#define KBLKS 32
#define NBLKS 32
#define NPER 32            // async b128 instructions per thread per stage (16 X + 16 W)

__device__ __forceinline__ uint32_t lds_off(const void* p) {
    // generic -> LDS address space; folds to the raw 32-bit LDS byte offset
    return (uint32_t)(uintptr_t)(__attribute__((address_space(3))) const void*)p;
}

__device__ __forceinline__ void async_b128(uint32_t lds_byte, const void* gsrc) {
    // GLOBAL_LOAD_ASYNC_TO_LDS_B128, GV mode: vdst = LDS byte addr, vaddr = 64-bit global
    asm volatile("global_load_async_to_lds_b128 %0, %1, off"
                 :: "v"(lds_byte), "v"((uint64_t)(uintptr_t)gsrc)
                 : "memory");
}

#define WAIT_ASYNC(n) asm volatile("s_wait_asynccnt %0" :: "i"(n) : "memory")

__global__ __launch_bounds__(256)
void mosaic_wmma_gemm(const float* __restrict__ x,
                      const float* __restrict__ tiles,
                      const float* __restrict__ bias,
                      const int*   __restrict__ parts,
                      float* __restrict__ y)
{
    __shared__ float x_lds[2][BM][XSTRIDE];   // X chunk, [m][k], double buffered
    __shared__ float w_lds[2][BK][BN];        // gathered tile, [k][n], double buffered

    const int tid  = threadIdx.x;
    const int lane = tid & 31;
    const int wv   = tid >> 5;     // wave 0..7
    const int hlf  = lane >> 4;    // half-wave 0/1
    const int lid  = lane & 15;

    const int nblk = blockIdx.x;   // 0..31
    const int mblk = blockIdx.y;   // 0..63

    const int wave_m = wv >> 2;    // 0..1  -> 64 rows
    const int wave_n = wv & 3;     // 0..3  -> 32 cols

    const size_t m_base = (size_t)mblk * BM;
    const size_t n_base = (size_t)nblk * BN;

    // One stage of async global->LDS copies; each instruction moves one full
    // 512B row across the wave (fully coalesced lines at L2).
    auto stage = [&](int kblk, int buf) {
        const float* xsrc = x + m_base * K_DIM + (size_t)kblk * BK;
        const int tile_id = parts[kblk * NBLKS + nblk];       // uniform -> s_load
        const float* wsrc = tiles + (size_t)tile_id * (128 * 128);
        const uint32_t xb = lds_off(&x_lds[buf][0][0]);
        const uint32_t wb = lds_off(&w_lds[buf][0][0]);
        #pragma unroll
        for (int i = 0; i < 16; ++i) {
            int idx  = tid + i * 256;          // 0..4095 float4s
            int row  = idx >> 5;               // 32 float4 per 128-float row
            int col4 = idx & 31;
            async_b128(xb + (uint32_t)(row * XSTRIDE + col4 * 4) * 4u,
                       xsrc + (size_t)row * K_DIM + col4 * 4);
        }
        #pragma unroll
        for (int i = 0; i < 16; ++i) {
            int idx  = tid + i * 256;
            int row  = idx >> 5;
            int col4 = idx & 31;
            async_b128(wb + (uint32_t)(row * BN + col4 * 4) * 4u,
                       wsrc + row * 128 + col4 * 4);
        }
    };

    v8f acc[4][2] = {};            // 4 M-subtiles x 2 N-subtiles of 16x16

    stage(0, 0);
    for (int kblk = 0; kblk < KBLKS; ++kblk) {
        const int buf = kblk & 1;
        if (kblk + 1 < KBLKS) {
            stage(kblk + 1, buf ^ 1);   // prefetch next stage into other buffer
            WAIT_ASYNC(NPER);           // in-order completion => stage kblk landed
        } else {
            WAIT_ASYNC(0);
        }
        __syncthreads();                // all waves' async writes visible

        // Hoisted fragment base pointers: every hot-loop DS access becomes
        // base + compile-time immediate (A max 25,844B; B max 64,064B < 64KB).
        const float* ab = &x_lds[buf][wave_m * 64 + lid][2 * hlf];
        const float* bb = &w_lds[buf][2 * hlf][wave_n * 32 + lid];

        #pragma unroll 4
        for (int k0 = 0; k0 < BK; k0 += 4) {
            // A frags: a[v] = X[m = lid][k = k0 + 2*hlf + v]   (ISA 16x4 f32 layout)
            v2f a[4];
            #pragma unroll
            for (int mi = 0; mi < 4; ++mi)
                a[mi] = *(const v2f*)(ab + mi * 16 * XSTRIDE + k0);
            // B frags: b[v] = W[k = k0 + 2*hlf + v][n = lid]   (4x16 layout)
            v2f b[2];
            #pragma unroll
            for (int ni = 0; ni < 2; ++ni) {
                b[ni].x = bb[(k0 + 0) * BN + ni * 16];
                b[ni].y = bb[(k0 + 1) * BN + ni * 16];
            }
            #pragma unroll
            for (int mi = 0; mi < 4; ++mi)
                #pragma unroll
                for (int ni = 0; ni < 2; ++ni)
                    acc[mi][ni] = __builtin_amdgcn_wmma_f32_16x16x4_f32(
                        false, a[mi], false, b[ni],
                        (short)0, acc[mi][ni], false, false);
        }
        __syncthreads();                // buffer free before stage kblk+2 writes it
    }

    // Epilogue: C/D layout is M = v + 8*hlf, N = lid; one base pointer per wave,
    // all stores fold to base + 24-bit immediate; nontemporal (write-once stream).
    {
        const size_t n0 = n_base + wave_n * 32 + lid;
        const float bv0 = bias[n0];
        const float bv1 = bias[n0 + 16];
        float* yb = y + (m_base + wave_m * 64 + 8 * hlf) * N_DIM + n0;
        #pragma unroll
        for (int mi = 0; mi < 4; ++mi)
            #pragma unroll
            for (int v = 0; v < 8; ++v) {
                __builtin_nontemporal_store(acc[mi][0][v] + bv0,
                                            yb + (size_t)(mi * 16 + v) * N_DIM);
                __builtin_nontemporal_store(acc[mi][1][v] + bv1,
                                            yb + (size_t)(mi * 16 + v) * N_DIM + 16);
            }
    }
}

extern "C" void kernel_launch(void* const* d_in, const int* in_sizes, int n_in,
                              void* d_out, int out_size, void* d_ws, size_t ws_size,
                              hipStream_t stream) {
    const float* x     = (const float*)d_in[0];
    const float* tiles = (const float*)d_in[1];
    const float* bias  = (const float*)d_in[2];
    const int*   parts = (const int*)d_in[3];
    float* y = (float*)d_out;

    const int M = in_sizes[0] / K_DIM;      // 8192
    dim3 grid(N_DIM / BN, M / BM);          // (32, 64)
    mosaic_wmma_gemm<<<grid, 256, 0, stream>>>(x, tiles, bias, parts, y);
}